// GradientCurvatureAttention_60748017434835
// MI455X (gfx1250) — compile-verified
//
#include <hip/hip_runtime.h>
#include <cstdint>

#define B_  16
#define C_  128
#define H_  128
#define W_  128
#define EPSV 1e-6f

// LDS geometry (floats)
#define XT_ROW   72     // staged row width: global cols [w0-4 .. w0+67]
#define XT_CH    220    // per-channel stride (3*72 padded, 16B aligned, bank-friendly)
#define S_STRIDE 65     // s[c][w] stride (odd -> bank friendly)

#if defined(__gfx1250__) && __has_builtin(__builtin_amdgcn_tensor_store_from_lds)
#define USE_TDM_STORE 1
typedef unsigned int u32x4 __attribute__((ext_vector_type(4)));
typedef int          i32x8 __attribute__((ext_vector_type(8)));
typedef int          i32x4 __attribute__((ext_vector_type(4)));
#else
#define USE_TDM_STORE 0
#endif

__device__ __forceinline__ void async_ld_b128(uint32_t lds_off, uint64_t gaddr) {
    // gfx1250: async DMA of 16B global -> LDS, tracked by ASYNCcnt
    asm volatile("global_load_async_to_lds_b128 %0, %1, off"
                 :: "v"(lds_off), "v"(gaddr) : "memory");
}
__device__ __forceinline__ void wait_async0() {
    asm volatile("s_wait_asynccnt 0" ::: "memory");
}
__device__ __forceinline__ void wait_tensor0() {
#if defined(__gfx1250__)
    asm volatile("s_wait_tensorcnt 0x0" ::: "memory");
#endif
}

__global__ __launch_bounds__(256)
void gca_kernel(const float* __restrict__ x, float* __restrict__ out) {
    __shared__ __align__(16) float lds_x[C_ * XT_CH];   // 112,640 B
    __shared__ float lds_s[C_ * S_STRIDE];              //  33,280 B
    __shared__ float lds_red[256];
    __shared__ float lds_red2[64];

    const int tid = threadIdx.x;
    const int idx = blockIdx.x;
    const int w0  = (idx & 1) * 64;
    const int h   = (idx >> 1) & (H_ - 1);
    const int b   = idx >> 8;

    // ---------------- Phase 1: async stage 3 rows x 128 channels ----------------
    const uint64_t xbase = (uint64_t)(uintptr_t)x;
    const uint32_t lx0   = (uint32_t)(uintptr_t)&lds_x[0];
    #pragma unroll
    for (int i = 0; i < 27; ++i) {                 // 6912 16B chunks / 256 threads
        int t  = tid + 256 * i;
        int k  = t % 18;                           // chunk within row
        int r  = (t / 18) % 3;                     // row 0..2 (h-1..h+1)
        int c  = t / 54;                           // channel
        int gr = h - 1 + r;
        int cs = w0 - 4 + 4 * k;                   // global start col of chunk
        int fi = c * XT_CH + r * XT_ROW + 4 * k;   // LDS float index
        if (gr >= 0 && gr < H_ && cs >= 0 && cs <= W_ - 4) {
            uint32_t goff = (uint32_t)((((b * C_ + c) * H_ + gr) * W_ + cs) * 4);
            async_ld_b128(lx0 + (uint32_t)(fi * 4), xbase + goff);
        } else {
            *reinterpret_cast<float4*>(&lds_x[fi]) = make_float4(0.f, 0.f, 0.f, 0.f);
        }
    }
    wait_async0();
    __syncthreads();

    // ---------------- Phase 2: separable stencils -> s[c][w] ----------------
    {
        const int c  = tid >> 1;
        const int B0 = (tid & 1) * 32;
        const int xb = c * XT_CH;
        float al, bl, cl2, am, bm, cm;
        {
            int j = B0 + 3;
            float r0 = lds_x[xb + j], r1 = lds_x[xb + XT_ROW + j], r2 = lds_x[xb + 2 * XT_ROW + j];
            float ts = r0 + r2;
            al = fmaf(2.f, r1, ts); bl = r0 - r2; cl2 = fmaf(-2.f, r1, ts);
            j = B0 + 4;
            r0 = lds_x[xb + j]; r1 = lds_x[xb + XT_ROW + j]; r2 = lds_x[xb + 2 * XT_ROW + j];
            ts = r0 + r2;
            am = fmaf(2.f, r1, ts); bm = r0 - r2; cm = fmaf(-2.f, r1, ts);
        }
        #pragma unroll 4
        for (int lw = 0; lw < 32; ++lw) {
            int j = B0 + 5 + lw;
            float r0 = lds_x[xb + j], r1 = lds_x[xb + XT_ROW + j], r2 = lds_x[xb + 2 * XT_ROW + j];
            float ts = r0 + r2;
            float ar = fmaf(2.f, r1, ts);
            float br = r0 - r2;
            float cr = fmaf(-2.f, r1, ts);
            // horizontal taps (cross-correlation, matches XLA conv)
            float Gx  = al - ar;
            float u   = al + ar;
            float Ixx = fmaf(-2.f, am, u);
            float v   = bl + br;
            float Gy  = fmaf(2.f, bm, v);
            float Ixy = br - bl;
            float w2  = cl2 + cr;
            float Iyy = fmaf(2.f, cm, w2);
            // elementwise
            float gx2 = Gx * Gx, gy2 = Gy * Gy;
            float g2  = gx2 + gy2 + EPSV;
            float sq  = sqrtf(g2);
            float num = gx2 * Iyy;
            num = fmaf(gy2, Ixx, num);
            float m2gxy = -2.f * Gx * Gy;
            num = fmaf(m2gxy, Ixy, num);
            float curv = num / (g2 * sq);
            lds_s[c * S_STRIDE + B0 + lw] = sq + curv;
            al = am; bl = bm; cl2 = cm; am = ar; bm = br; cm = cr;
        }
    }
    __syncthreads();

    // ---------------- Phase 3: softmax over C in LDS ----------------
    const int p  = tid >> 6;   // 0..3, 32 channels each
    const int wl = tid & 63;   // local column
    {
        float m = -1e30f;
        int cb = p * 32;
        #pragma unroll 8
        for (int i = 0; i < 32; ++i)
            m = fmaxf(m, lds_s[(cb + i) * S_STRIDE + wl]);
        lds_red[p * 64 + wl] = m;
    }
    __syncthreads();
    if (tid < 64) {
        float m = fmaxf(fmaxf(lds_red[wl], lds_red[64 + wl]),
                        fmaxf(lds_red[128 + wl], lds_red[192 + wl]));
        lds_red2[wl] = m;
    }
    __syncthreads();
    {
        float gm  = lds_red2[wl];
        float acc = 0.f;
        int cb = p * 32;
        #pragma unroll 8
        for (int i = 0; i < 32; ++i) {
            int si  = (cb + i) * S_STRIDE + wl;
            float e = __expf(lds_s[si] - gm);
            lds_s[si] = e;
            acc += e;
        }
        lds_red[p * 64 + wl] = acc;
    }
    __syncthreads();
    if (tid < 64) {
        float ssum = lds_red[wl] + lds_red[64 + wl] + lds_red[128 + wl] + lds_red[192 + wl];
        lds_red2[wl] = 1.0f / ssum;
    }
    __syncthreads();

    // ---------------- Phase 4: out = (softmax + 1) * x ----------------
#if USE_TDM_STORE
    {
        float inv = lds_red2[wl];
        float xc[32];
        #pragma unroll
        for (int i = 0; i < 32; ++i) {           // cache center-row x in registers
            int c = p + (i << 2);
            xc[i] = lds_x[c * XT_CH + XT_ROW + (wl + 4)];
        }
        __syncthreads();                          // all lds_x reads done before clobber
        #pragma unroll
        for (int i = 0; i < 32; ++i) {            // pack results contiguously: [c][64]
            int c = p + (i << 2);
            float e = lds_s[c * S_STRIDE + wl];
            lds_x[c * 64 + wl] = fmaf(e, inv, 1.0f) * xc[i];
        }
    }
    __syncthreads();
    if (tid < 32) {                               // wave 0 issues one TDM store (EXEC-agnostic op)
        const uint64_t gout = (uint64_t)(uintptr_t)out +
            (uint64_t)((((size_t)b * C_) * H_ + h) * W_ + w0) * 4u;
        // ---- D# group 0: count=1 | lds_addr | global_addr(57b) | type=2
        u32x4 g0;
        g0.x = 1u;
        g0.y = lx0;                               // LDS byte address of packed results
        g0.z = (uint32_t)gout;
        g0.w = (uint32_t)((gout >> 32) & 0x01FFFFFFu) | (2u << 30);
        // ---- D# group 1: data_size=4B; 2D tile 64 x 128; line stride = H*W
        const uint32_t td0 = (uint32_t)(W_ - w0); // valid cols from tile start
        const uint32_t td1 = (uint32_t)C_;        // valid lines (channels)
        const uint64_t st0 = (uint64_t)(H_ * W_); // elements between lines
        i32x8 g1;
        g1[0] = 0x00020000;                                        // data_size=2 (4B)
        g1[1] = (int)((td0 & 0xFFFFu) << 16);                      // tensor_dim0[15:0]
        g1[2] = (int)(((td0 >> 16) & 0xFFFFu) | ((td1 & 0xFFFFu) << 16));
        g1[3] = (int)(((td1 >> 16) & 0xFFFFu) | (64u << 16));      // tile_dim0 = 64
        g1[4] = (int)(128u);                                       // tile_dim1=128, tile_dim2=0
        g1[5] = (int)(uint32_t)(st0 & 0xFFFFFFFFu);                // dim0 stride lo
        g1[6] = (int)(uint32_t)((st0 >> 32) & 0xFFFFu);            // dim0 stride hi, dim1 stride=0
        g1[7] = 0;
        i32x4 g2 = {0x40000000, 0, 0, 0};         // protective tensor_dim2 (dim2 unused)
        i32x4 g3 = {0, 0, 0, 0};
        i32x8 g4 = {0, 0, 0, 0, 0, 0, 0, 0};      // unused 5th operand slot (VADDR4 = NULL per ISA)
        __builtin_amdgcn_tensor_store_from_lds(g0, g1, g2, g3, g4, 0);
        wait_tensor0();                           // drain TENSORcnt (ENDPGM would too)
    }
#else
    {
        float inv = lds_red2[wl];
        #pragma unroll 4
        for (int i = 0; i < 32; ++i) {
            int c = p + (i << 2);
            float e    = lds_s[c * S_STRIDE + wl];
            float attn = fmaf(e, inv, 1.0f);
            float xc   = lds_x[c * XT_CH + XT_ROW + (wl + 4)];
            out[(((size_t)b * C_ + c) * H_ + h) * W_ + (w0 + wl)] = attn * xc;
        }
    }
#endif
}

extern "C" void kernel_launch(void* const* d_in, const int* in_sizes, int n_in,
                              void* d_out, int out_size, void* d_ws, size_t ws_size,
                              hipStream_t stream) {
    const float* x = (const float*)d_in[0];
    float* out     = (float*)d_out;
    dim3 grid(B_ * H_ * 2);   // (b, h, w-half) -> 4096 blocks
    dim3 block(256);          // 8 wave32s
    hipLaunchKernelGGL(gca_kernel, grid, block, 0, stream, x, out);
}